// FocalLoss_79963701117050
// MI455X (gfx1250) — compile-verified
//
#include <hip/hip_runtime.h>
#include <math.h>

// -----------------------------------------------------------------------------
// Focal loss on MI455X (gfx1250, wave32).
//
// Stage 1 (bandwidth-bound, the 262 MB pass):
//   one wave32 per row of 1000 fp32 logits. Row is loaded with coalesced
//   global_load_b128 (250 float4 chunks, lane-strided) and kept register-
//   resident (8 float4 = 32 VGPRs/lane), so max and sum(exp) are computed in a
//   SINGLE HBM pass -- crucial because 262 MB > 192 MB L2, a second pass would
//   hit HBM again. Wave-level shuffle allreduces for max / sum.
//
// Stage 2 (exact fp32 WMMA reduction):
//   65536 row losses -> 256 partials (fixed order) -> viewed as a 16x16 fp32
//   A-matrix, reduced with V_WMMA_F32_16X16X4_F32 against an all-ones B
//   (4 accumulating WMMAs, k = 0..15). Full-precision fp32 matrix op, fixed
//   operation order => deterministic across graph replays (no float atomics).
// -----------------------------------------------------------------------------

#define CLS      1000
#define CHUNKS4  250            // CLS / 4 float4 chunks per row
#define ROWS_PER_BLOCK 8        // 256 threads = 8 wave32, one row per wave

typedef __attribute__((ext_vector_type(2))) float v2f;
typedef __attribute__((ext_vector_type(8))) float v8f;

__global__ __launch_bounds__(256) void focal_rows_kernel(
    const float* __restrict__ x,       // [N, CLS]
    const int*   __restrict__ tgt,     // [N]
    const float* __restrict__ gtab,    // [CLS]
    const float* __restrict__ atab,    // [CLS]
    float*       __restrict__ row_loss,// [N] (workspace)
    int n_rows)
{
    const int lane = threadIdx.x & 31;
    const int wave = threadIdx.x >> 5;
    const int row  = blockIdx.x * ROWS_PER_BLOCK + wave;
    if (row >= n_rows) return;

    const float* xrow = x + (size_t)row * CLS;                 // 16B aligned (4000 % 16 == 0)
    const float4* xr4 = reinterpret_cast<const float4*>(xrow);

    // ---- single coalesced load pass: row -> registers -------------------
    float4 v[8];
#pragma unroll
    for (int j = 0; j < 7; ++j)            // chunks 0..223: full for all lanes
        v[j] = xr4[lane + 32 * j];
    {
        const int c = lane + 224;          // tail: chunks 224..249 (lanes 0..25)
        if (c < CHUNKS4) v[7] = xr4[c];
        else             v[7] = make_float4(-INFINITY, -INFINITY, -INFINITY, -INFINITY);
    }

    // ---- per-row max (registers + wave32 allreduce) ---------------------
    float m = -INFINITY;
#pragma unroll
    for (int j = 0; j < 8; ++j)
        m = fmaxf(m, fmaxf(fmaxf(v[j].x, v[j].y), fmaxf(v[j].z, v[j].w)));
#pragma unroll
    for (int off = 16; off >= 1; off >>= 1)
        m = fmaxf(m, __shfl_xor(m, off, 32));

    // ---- sum of exp(x - max); exp(-inf)=0 handles tail lanes ------------
    float s = 0.0f;
#pragma unroll
    for (int j = 0; j < 8; ++j) {
        s += __expf(v[j].x - m);
        s += __expf(v[j].y - m);
        s += __expf(v[j].z - m);
        s += __expf(v[j].w - m);
    }
#pragma unroll
    for (int off = 16; off >= 1; off >>= 1)
        s += __shfl_xor(s, off, 32);

    // ---- focal epilogue (one lane per row) ------------------------------
    if (lane == 0) {
        const int   t  = tgt[row];
        const float xt = xrow[t];
        const float ce = m + __logf(s) - xt;        // -log_softmax[target]
        const float pt = __expf(-ce);               // p_t
        const float g  = gtab[t];
        const float a  = atab[t];
        row_loss[row]  = a * powf(1.0f - pt, g) * ce;
    }
}

__global__ __launch_bounds__(256) void reduce_mean_wmma_kernel(
    const float* __restrict__ row_loss, float* __restrict__ out, int n)
{
    __shared__ float lds[256];
    const int t = threadIdx.x;

    // 65536 -> 256 partials, fixed order, coalesced strided reads.
    float s = 0.0f;
    for (int j = t; j < n; j += 256) s += row_loss[j];
    lds[t] = s;
    __syncthreads();

    // Wave 0 only (wave-uniform branch => EXEC all-ones inside, as WMMA needs).
    if (t < 32) {
        const int lane = t;
        // View lds[256] as fp32 A[16][16] with A[m][k] = lds[m*16 + k].
        // V_WMMA_F32_16X16X4_F32 A-layout: lanes 0-15 hold (M=lane, K=0,1),
        // lanes 16-31 hold (M=lane-16, K=2,3). B = all-ones (layout invariant),
        // so D[m][*] = row sum of A; accumulate 4 chunks to cover K=0..15.
        v8f acc = {};
#pragma unroll
        for (int q = 0; q < 4; ++q) {
            const int mrow = lane & 15;
            const int k0   = 4 * q + ((lane < 16) ? 0 : 2);
            v2f a; a[0] = lds[mrow * 16 + k0]; a[1] = lds[mrow * 16 + k0 + 1];
            v2f b; b[0] = 1.0f;                b[1] = 1.0f;
            acc = __builtin_amdgcn_wmma_f32_16x16x4_f32(
                /*neg_a=*/false, a, /*neg_b=*/false, b,
                /*c_mod=*/(short)0, acc, /*reuse_a=*/false, /*reuse_b=*/false);
        }
        // D layout: VGPR v holds M=v (lanes 0-15) / M=v+8 (lanes 16-31).
        // Sum of 8 acc regs => half-total per lane half; combine halves.
        float tot = acc[0] + acc[1] + acc[2] + acc[3] +
                    acc[4] + acc[5] + acc[6] + acc[7];
        tot += __shfl_xor(tot, 16, 32);
        if (lane == 0) out[0] = tot / (float)n;
    }
}

extern "C" void kernel_launch(void* const* d_in, const int* in_sizes, int n_in,
                              void* d_out, int out_size, void* d_ws, size_t ws_size,
                              hipStream_t stream) {
    const float* x    = (const float*)d_in[0];  // [N, 1000] fp32
    const int*   tgt  = (const int*)  d_in[1];  // [N] int
    const float* gtab = (const float*)d_in[2];  // [1000] fp32
    const float* atab = (const float*)d_in[3];  // [1000] fp32
    float* out      = (float*)d_out;            // [1] fp32
    float* row_loss = (float*)d_ws;             // [N] fp32 scratch (256 KB)

    const int n = in_sizes[1];                  // 65536 rows
    const int blocks = (n + ROWS_PER_BLOCK - 1) / ROWS_PER_BLOCK;

    focal_rows_kernel<<<blocks, 256, 0, stream>>>(x, tgt, gtab, atab, row_loss, n);
    reduce_mean_wmma_kernel<<<1, 256, 0, stream>>>(row_loss, out, n);
}